// RNNModel_29308856828070
// MI455X (gfx1250) — compile-verified
//
#include <hip/hip_runtime.h>
#include <stddef.h>

// ---------------------------------------------------------------------------
// tanh: CDNA5 hardware v_tanh_f32 (confirmed lowering via probe in round 3) —
// a single TRANS op on the serial critical path. Guarded fallback kept for
// portability: tanh(v) = 1 - 2/(exp(2v)+1) via v_exp_f32 + v_rcp_f32.
// ---------------------------------------------------------------------------
#if defined(__HIP_DEVICE_COMPILE__) && defined(__gfx1250__) && __has_builtin(__builtin_amdgcn_tanhf)
__device__ __forceinline__ float fast_tanh(float v) {
    return __builtin_amdgcn_tanhf(v);          // -> v_tanh_f32_e32
}
#else
__device__ __forceinline__ float fast_tanh(float v) {
    float e = __expf(2.0f * v);                // v_exp_f32 (with log2e scale)
    return 1.0f - 2.0f * __builtin_amdgcn_rcpf(e + 1.0f);
}
#endif

// ---------------------------------------------------------------------------
// Main RNN scan. One lane owns one batch element. H==2, I==1, O==1 hardwired.
//   h = tanh(W_ih*x_t + b_ih + W_hh*h + b_hh), frozen past lengths[b].
//   out[b] = fc_w . h + fc_b
// Per-step critical path: v_fma -> v_fma -> v_tanh (minimum for H=2 f32).
// float4 streaming loads, register double-buffering, global_prefetch ahead.
// ---------------------------------------------------------------------------
__device__ __forceinline__ void rnn_step(float& h0, float& h1,
                                         float p0, float p1,
                                         float a00, float a01,
                                         float a10, float a11) {
    // p0/p1 = x_t*w + (b_ih+b_hh), precomputed off the critical chain.
    float u0 = fmaf(h1, a01, fmaf(h0, a00, p0));
    float u1 = fmaf(h1, a11, fmaf(h0, a10, p1));
    h0 = fast_tanh(u0);
    h1 = fast_tanh(u1);
}

__global__ __launch_bounds__(64) void rnn_scan_kernel(
    const float* __restrict__ x,        // [B, T, 1]
    const int*   __restrict__ lengths,  // [B]
    const float* __restrict__ W_ih,     // [2, 1]
    const float* __restrict__ W_hh,     // [2, 2]
    const float* __restrict__ b_ih,     // [2]
    const float* __restrict__ b_hh,     // [2]
    const float* __restrict__ fc_w,     // [1, 2]
    const float* __restrict__ fc_b,     // [1]
    const int*   __restrict__ perm,     // [B] or nullptr (identity)
    float*       __restrict__ out,      // [B, 1]
    int B, int T)
{
    const int i = blockIdx.x * blockDim.x + threadIdx.x;
    if (i >= B) return;
    const int b = perm ? perm[i] : i;

    // Uniform weights -> scalar loads (s_load_b512), broadcast to the wave.
    const float w0  = W_ih[0], w1  = W_ih[1];
    const float a00 = W_hh[0], a01 = W_hh[1];
    const float a10 = W_hh[2], a11 = W_hh[3];
    const float c0  = b_ih[0] + b_hh[0];
    const float c1  = b_ih[1] + b_hh[1];

    const int len = lengths[b];
    const float* __restrict__ xp = x + (size_t)b * (size_t)T;

    float h0 = 0.0f, h1 = 0.0f;
    int t = 0;

    if ((T & 3) == 0) {                       // 16B-aligned float4 path
        const int len4 = len & ~3;
        if (len4 > 0) {
            float4 cur = *reinterpret_cast<const float4*>(xp);
            for (t = 0; t + 4 < len4; t += 4) {
                // Next chunk's load issues before this chunk's tanh chain.
                float4 nxt = *reinterpret_cast<const float4*>(xp + t + 4);
                __builtin_prefetch(xp + t + 132, 0, 1);   // ~512B ahead -> global_prefetch_b8
                rnn_step(h0, h1, fmaf(cur.x, w0, c0), fmaf(cur.x, w1, c1), a00, a01, a10, a11);
                rnn_step(h0, h1, fmaf(cur.y, w0, c0), fmaf(cur.y, w1, c1), a00, a01, a10, a11);
                rnn_step(h0, h1, fmaf(cur.z, w0, c0), fmaf(cur.z, w1, c1), a00, a01, a10, a11);
                rnn_step(h0, h1, fmaf(cur.w, w0, c0), fmaf(cur.w, w1, c1), a00, a01, a10, a11);
                cur = nxt;
            }
            rnn_step(h0, h1, fmaf(cur.x, w0, c0), fmaf(cur.x, w1, c1), a00, a01, a10, a11);
            rnn_step(h0, h1, fmaf(cur.y, w0, c0), fmaf(cur.y, w1, c1), a00, a01, a10, a11);
            rnn_step(h0, h1, fmaf(cur.z, w0, c0), fmaf(cur.z, w1, c1), a00, a01, a10, a11);
            rnn_step(h0, h1, fmaf(cur.w, w0, c0), fmaf(cur.w, w1, c1), a00, a01, a10, a11);
            t += 4;                            // t == len4
        }
    }
    for (; t < len; ++t) {                     // scalar tail (len % 4 steps)
        float xt = xp[t];
        rnn_step(h0, h1, fmaf(xt, w0, c0), fmaf(xt, w1, c1), a00, a01, a10, a11);
    }

    out[b] = fmaf(h1, fc_w[1], fmaf(h0, fc_w[0], fc_b[0]));
}

// ---------------------------------------------------------------------------
// Length-sorting helpers (counting sort over length bins [0, T]).
// perm[i] = batch index of the i-th shortest sequence; lanes get similar
// lengths -> minimal intra-wave divergence in the serial scan.
// ---------------------------------------------------------------------------
__global__ void rnn_zero_counts(int* __restrict__ counts, int n) {
    int i = blockIdx.x * blockDim.x + threadIdx.x;
    if (i < n) counts[i] = 0;
}

__global__ void rnn_hist_lengths(const int* __restrict__ lengths,
                                 int* __restrict__ counts, int B, int T) {
    int b = blockIdx.x * blockDim.x + threadIdx.x;
    if (b < B) {
        int l = lengths[b];
        l = (l < 0) ? 0 : (l > T ? T : l);
        atomicAdd(&counts[l], 1);
    }
}

__global__ void rnn_scan_counts(int* __restrict__ counts, int n) {
    if (blockIdx.x == 0 && threadIdx.x == 0) {
        int acc = 0;
        for (int i = 0; i < n; ++i) {   // n = T+1 = 2049 serial adds: negligible
            int c = counts[i];
            counts[i] = acc;
            acc += c;
        }
    }
}

__global__ void rnn_scatter_perm(const int* __restrict__ lengths,
                                 int* __restrict__ counts,
                                 int* __restrict__ perm, int B, int T) {
    int b = blockIdx.x * blockDim.x + threadIdx.x;
    if (b < B) {
        int l = lengths[b];
        l = (l < 0) ? 0 : (l > T ? T : l);
        int pos = atomicAdd(&counts[l], 1);
        perm[pos] = b;
    }
}

// ---------------------------------------------------------------------------
// Launcher
// ---------------------------------------------------------------------------
extern "C" void kernel_launch(void* const* d_in, const int* in_sizes, int n_in,
                              void* d_out, int out_size, void* d_ws, size_t ws_size,
                              hipStream_t stream) {
    (void)n_in; (void)out_size;
    const float* x       = (const float*)d_in[0];
    const int*   lengths = (const int*)  d_in[1];
    const float* W_ih    = (const float*)d_in[2];
    const float* W_hh    = (const float*)d_in[3];
    const float* b_ih    = (const float*)d_in[4];
    const float* b_hh    = (const float*)d_in[5];
    const float* fc_w    = (const float*)d_in[6];
    const float* fc_b    = (const float*)d_in[7];
    float*       out     = (float*)d_out;

    const int B = in_sizes[1];                 // lengths: [B]
    const int T = (B > 0) ? (in_sizes[0] / B) : 0;   // x: [B, T, 1]
    if (B <= 0 || T <= 0) return;

    // Workspace: counts[T+1] then perm[B]
    int* counts = (int*)d_ws;
    int* perm   = counts + (T + 1);
    const bool sort_ok =
        (d_ws != nullptr) && (ws_size >= (size_t)(T + 1 + B) * sizeof(int));

    if (sort_ok) {
        int nbins = T + 1;
        rnn_zero_counts <<<(nbins + 255) / 256, 256, 0, stream>>>(counts, nbins);
        rnn_hist_lengths<<<(B + 255) / 256,     256, 0, stream>>>(lengths, counts, B, T);
        rnn_scan_counts <<<1, 1, 0, stream>>>(counts, nbins);
        rnn_scatter_perm<<<(B + 255) / 256,     256, 0, stream>>>(lengths, counts, perm, B, T);
    }

    const int BLK = 64;                        // 2 waves/WG, 128 WGs -> spread wide
    rnn_scan_kernel<<<(B + BLK - 1) / BLK, BLK, 0, stream>>>(
        x, lengths, W_ih, W_hh, b_ih, b_hh, fc_w, fc_b,
        sort_ok ? perm : nullptr, out, B, T);
}